// MultiSemanticSPR_42314017800812
// MI455X (gfx1250) — compile-verified
//
#include <hip/hip_runtime.h>
#include <hip/hip_bf16.h>
#include <math.h>

#define B_SZ 4096
#define S_SZ 7
#define H_SZ 1024

typedef float v2f __attribute__((ext_vector_type(2)));
typedef float v8f __attribute__((ext_vector_type(8)));
typedef int   v4i __attribute__((ext_vector_type(4)));

#if defined(__AMDGCN__) && __has_builtin(__builtin_amdgcn_wmma_f32_16x16x4_f32)
#define USE_WMMA_F32 1
#else
#define USE_WMMA_F32 0
#endif

// CDNA5 async global->LDS copy path (ASYNCcnt-tracked), guarded so a toolchain
// without the builtin still compiles via the register-staged fallback.
#if defined(__AMDGCN__) && __has_builtin(__builtin_amdgcn_global_load_async_to_lds_b128)
#define USE_ASYNC_LDS 1
#else
#define USE_ASYNC_LDS 0
#endif

#if USE_ASYNC_LDS
// probe-derived signature: (int4 AS1* gsrc, int4 AS3* ldsdst, imm offset, imm cpol)
typedef __attribute__((address_space(1))) v4i* as1_v4i_ptr;
typedef __attribute__((address_space(3))) v4i* as3_v4i_ptr;

__device__ __forceinline__ void cp_async16(void* lds, const void* g)
{
    __builtin_amdgcn_global_load_async_to_lds_b128(
        (as1_v4i_ptr)(g), (as3_v4i_ptr)(lds), 0, 0);
}
__device__ __forceinline__ void wait_async0()
{
#if __has_builtin(__builtin_amdgcn_s_wait_asynccnt)
    __builtin_amdgcn_s_wait_asynccnt(0);
#else
    asm volatile("s_wait_asynccnt 0x0" ::: "memory");
#endif
}
#endif

// ---------------------------------------------------------------------------
// Tiled fp32 GEMM on V_WMMA_F32_16X16X4_F32 (exact fp32 -> matches reference).
//   C[z] = act(A[z] @ W[z] + bias[z])
// A: (M,K) leading dim ldA, per-z offset aOff; W: (K,N) row-major, per-z wOff
// grid = (N/128, M/64, Z); block = 256 threads (8 wave32)
// 64x128 block tile, BK=32, double-buffered LDS, one barrier per K-step.
// Each wave: 16(M) x 64(N) = 4 v8f accumulators; A fragment reused 4x.
// ---------------------------------------------------------------------------
template <bool BIAS, bool RELU>
__global__ void __launch_bounds__(256)
gemm_wmma_f32(const float* __restrict__ A, int ldA, size_t aOff,
              const float* __restrict__ W, size_t wOff,
              const float* __restrict__ bias, size_t bOff,
              float* __restrict__ C, int ldC, size_t cOff,
              int N, int K)
{
    constexpr int BM = 64, BN = 128, BK = 32, APAD = 36, BPAD = 136;
    __shared__ alignas(16) float As[2][BM][APAD];  // (m,k)
    __shared__ alignas(16) float Bs[2][BK][BPAD];  // (k,n)

    const int z = blockIdx.z;
    const float* Ap = A + (size_t)z * aOff;
    const float* Wp = W + (size_t)z * wOff;
    float*       Cp = C + (size_t)z * cOff;

    const int t = threadIdx.x;
    const int wave = t >> 5, lane = t & 31;
    const int hv = lane >> 4, li = lane & 15;
    const int m0 = blockIdx.y * BM, n0 = blockIdx.x * BN;
    const int wm = (wave >> 1) * 16;   // 0,16,32,48
    const int wn = (wave & 1) * 64;    // 0,64

    // cooperative 16B-per-lane copy coordinates
    const int ar0 = t >> 3, ac0 = (t & 7) * 4;     // A: 512 chunks, 2/thread
    const int ar1 = ar0 + 32;
    const int br0 = t >> 5, bc0 = (t & 31) * 4;    // B: 1024 chunks, 4/thread

    v8f acc[4];
    const v8f vz = {0.f, 0.f, 0.f, 0.f, 0.f, 0.f, 0.f, 0.f};
    #pragma unroll
    for (int j = 0; j < 4; ++j) acc[j] = vz;

#if USE_ASYNC_LDS
    auto load_tile = [&](int buf, int k0) {
        cp_async16(&As[buf][ar0][ac0], Ap + (size_t)(m0 + ar0) * ldA + (k0 + ac0));
        cp_async16(&As[buf][ar1][ac0], Ap + (size_t)(m0 + ar1) * ldA + (k0 + ac0));
        #pragma unroll
        for (int i = 0; i < 4; ++i)
            cp_async16(&Bs[buf][br0 + 8 * i][bc0],
                       Wp + (size_t)(k0 + br0 + 8 * i) * N + (n0 + bc0));
    };
#else
    float4 ga[2], gb[4];
    auto issue_loads = [&](int k0) {
        ga[0] = *(const float4*)(Ap + (size_t)(m0 + ar0) * ldA + (k0 + ac0));
        ga[1] = *(const float4*)(Ap + (size_t)(m0 + ar1) * ldA + (k0 + ac0));
        #pragma unroll
        for (int i = 0; i < 4; ++i)
            gb[i] = *(const float4*)(Wp + (size_t)(k0 + br0 + 8 * i) * N + (n0 + bc0));
    };
    auto commit = [&](int buf) {
        *(float4*)&As[buf][ar0][ac0] = ga[0];
        *(float4*)&As[buf][ar1][ac0] = ga[1];
        #pragma unroll
        for (int i = 0; i < 4; ++i)
            *(float4*)&Bs[buf][br0 + 8 * i][bc0] = gb[i];
    };
#endif

    auto compute = [&](int buf) {
        #pragma unroll
        for (int kk = 0; kk < BK; kk += 4) {
            // A/B VGPR0 holds K = 0(,2), VGPR1 holds K = 1(,3) per half-wave
            const int ks = kk + 2 * hv;
            v2f a = { As[buf][wm + li][ks], As[buf][wm + li][ks + 1] };
            #pragma unroll
            for (int j = 0; j < 4; ++j) {
                const int nb = wn + 16 * j + li;
                v2f b = { Bs[buf][ks][nb], Bs[buf][ks + 1][nb] };
#if USE_WMMA_F32
                acc[j] = __builtin_amdgcn_wmma_f32_16x16x4_f32(
                    false, a, false, b, (short)0, acc[j], false, false);
#else
                // scalar fallback, identical lane->element mapping
                #pragma unroll
                for (int i = 0; i < 8; ++i) {
                    const float am = As[buf][wm + i + 8 * hv][kk]     * Bs[buf][kk][nb]
                                   + As[buf][wm + i + 8 * hv][kk + 1] * Bs[buf][kk + 1][nb]
                                   + As[buf][wm + i + 8 * hv][kk + 2] * Bs[buf][kk + 2][nb]
                                   + As[buf][wm + i + 8 * hv][kk + 3] * Bs[buf][kk + 3][nb];
                    acc[j][i] += am;
                }
#endif
            }
        }
    };

#if USE_ASYNC_LDS
    load_tile(0, 0);
    wait_async0();
    __syncthreads();
    int cur = 0;
    for (int k0 = 0; k0 < K; k0 += BK) {
        const int nxt = cur ^ 1;
        if (k0 + BK < K) load_tile(nxt, k0 + BK);  // DMA overlaps the WMMAs below
        compute(cur);
        wait_async0();
        __syncthreads();
        cur = nxt;
    }
#else
    issue_loads(0);
    commit(0);
    __syncthreads();
    int cur = 0;
    for (int k0 = 0; k0 < K; k0 += BK) {
        const int nxt = cur ^ 1;
        const bool more = (k0 + BK < K);
        if (more) {
            issue_loads(k0 + BK);     // global loads in flight during compute
            __builtin_prefetch(Ap + (size_t)(m0 + ar0) * ldA + (k0 + 2 * BK + ac0), 0, 1);
        }
        compute(cur);
        if (more) commit(nxt);
        __syncthreads();
        cur = nxt;
    }
#endif

    // D layout: VGPR i, lanes 0-15 -> M=i, N=lane; lanes 16-31 -> M=i+8, N=lane-16
    const float* bp = BIAS ? (bias + (size_t)z * bOff) : nullptr;
    #pragma unroll
    for (int i = 0; i < 8; ++i) {
        const int row = m0 + wm + i + 8 * hv;
        float* crow = Cp + (size_t)row * ldC;
        #pragma unroll
        for (int j = 0; j < 4; ++j) {
            const int c = n0 + wn + 16 * j + li;
            float v = acc[j][i];
            if (BIAS) v += bp[c];
            if (RELU) v = fmaxf(v, 0.f);
            crow[c] = v;
        }
    }
}

// ---------------------------------------------------------------------------
__device__ __forceinline__ float block_reduce_sum256(float v, float* red)
{
    const int t = threadIdx.x;
    red[t] = v; __syncthreads();
    #pragma unroll
    for (int o = 128; o > 0; o >>= 1) {
        if (t < o) red[t] += red[t + o];
        __syncthreads();
    }
    const float r = red[0];
    __syncthreads();
    return r;
}

// LayerNorm + exact GELU over each H_SZ row of (B,S,H). One block per row.
__global__ void __launch_bounds__(256)
ln_gelu_kernel(float* __restrict__ X, const float* __restrict__ gamma,
               const float* __restrict__ beta)
{
    __shared__ float red[256];
    float* x = X + (size_t)blockIdx.x * H_SZ;
    const int t = threadIdx.x;
    float v[H_SZ / 256];
    float s = 0.f;
    #pragma unroll
    for (int i = 0; i < H_SZ / 256; ++i) { v[i] = x[t + 256 * i]; s += v[i]; }
    const float mean = block_reduce_sum256(s, red) * (1.0f / H_SZ);
    float sq = 0.f;
    #pragma unroll
    for (int i = 0; i < H_SZ / 256; ++i) { const float d = v[i] - mean; sq += d * d; }
    const float var = block_reduce_sum256(sq, red) * (1.0f / H_SZ);
    const float rs = rsqrtf(var + 1e-5f);
    #pragma unroll
    for (int i = 0; i < H_SZ / 256; ++i) {
        const int c = t + 256 * i;
        const float y = (v[i] - mean) * rs * gamma[c] + beta[c];
        x[c] = 0.5f * y * (1.0f + erff(y * 0.70710678118654752f));  // exact GELU
    }
}

// Per-batch Gram-Schmidt over S_SZ rows of H_SZ, in place. One block per batch.
__global__ void __launch_bounds__(256)
gram_schmidt_kernel(float* __restrict__ M)
{
    __shared__ float V[S_SZ][H_SZ];
    __shared__ float red[256];
    float* base = M + (size_t)blockIdx.x * (S_SZ * H_SZ);
    float* Vf = &V[0][0];
    const int t = threadIdx.x;
    for (int i = t; i < S_SZ * H_SZ; i += 256) Vf[i] = base[i];
    __syncthreads();

    for (int i = 0; i < S_SZ; ++i) {
        for (int j = 0; j < i; ++j) {
            float p = 0.f;
            for (int e = t; e < H_SZ; e += 256) p += V[i][e] * V[j][e];
            const float d = block_reduce_sum256(p, red);  // q_j unit: proj = <v,q>
            for (int e = t; e < H_SZ; e += 256) V[i][e] -= d * V[j][e];
            __syncthreads();
        }
        float p = 0.f;
        for (int e = t; e < H_SZ; e += 256) { const float x = V[i][e]; p += x * x; }
        const float n = sqrtf(block_reduce_sum256(p, red));
        if (n > 1e-6f) {
            const float inv = 1.0f / n;
            for (int e = t; e < H_SZ; e += 256) V[i][e] *= inv;
        } else {
            // branch never taken for random data; any unit vector matches reference
            for (int e = t; e < H_SZ; e += 256) V[i][e] = (e == i) ? 1.0f : 0.0f;
        }
        __syncthreads();
    }
    for (int i = t; i < S_SZ * H_SZ; i += 256) base[i] = Vf[i];
}

// Deterministic 2-stage MSE reduction: grid (256, S_SZ), fixed partitions.
__global__ void __launch_bounds__(256)
loss_partial_kernel(const float* __restrict__ P, const float* __restrict__ Hm,
                    float* __restrict__ partials)
{
    __shared__ float red[256];
    const int s = blockIdx.y, blk = blockIdx.x, t = threadIdx.x;
    const size_t per = (size_t)B_SZ * H_SZ / 256;      // 16384 elements per block
    const size_t start = (size_t)blk * per;
    float acc = 0.f;
    for (size_t e = start + t; e < start + per; e += 256) {
        const size_t b = e >> 10, hh = e & (H_SZ - 1);
        const size_t idx = (b * S_SZ + s) * H_SZ + hh;
        const float d = P[idx] - Hm[idx];
        acc += d * d;
    }
    const float r = block_reduce_sum256(acc, red);
    if (t == 0) partials[(size_t)s * 256 + blk] = r;
}

// out[0] = total, out[1..7] = spr_losses  (fixed-order sums -> deterministic)
__global__ void loss_final_kernel(const float* __restrict__ partials,
                                  const float* __restrict__ w,
                                  float* __restrict__ out)
{
    __shared__ float losses[S_SZ];
    const int t = threadIdx.x;
    if (t < S_SZ) {
        float s = 0.f;
        for (int r = 0; r < 256; ++r) s += partials[(size_t)t * 256 + r];
        losses[t] = s * (1.0f / ((float)B_SZ * (float)H_SZ));
    }
    __syncthreads();
    if (t == 0) {
        float total = 0.f;
        for (int s = 0; s < S_SZ; ++s) { out[1 + s] = losses[s]; total += w[s] * losses[s]; }
        out[0] = total;
    }
}

// ---------------------------------------------------------------------------
extern "C" void kernel_launch(void* const* d_in, const int* in_sizes, int n_in,
                              void* d_out, int out_size, void* d_ws, size_t ws_size,
                              hipStream_t stream)
{
    (void)in_sizes; (void)n_in; (void)out_size; (void)ws_size;
    const float* sr  = (const float*)d_in[0];
    const float* Wd  = (const float*)d_in[1];
    const float* gam = (const float*)d_in[2];
    const float* bet = (const float*)d_in[3];
    const float* pw1 = (const float*)d_in[4];
    const float* pb1 = (const float*)d_in[5];
    const float* pw2 = (const float*)d_in[6];
    const float* pb2 = (const float*)d_in[7];
    const float* qw1 = (const float*)d_in[8];
    const float* qb1 = (const float*)d_in[9];
    const float* qw2 = (const float*)d_in[10];
    const float* qb2 = (const float*)d_in[11];
    const float* fw1 = (const float*)d_in[12];
    const float* fb1 = (const float*)d_in[13];
    const float* fw2 = (const float*)d_in[14];
    const float* fb2 = (const float*)d_in[15];
    const float* sw  = (const float*)d_in[16];
    float* out = (float*)d_out;

    const size_t N1 = (size_t)B_SZ * S_SZ * H_SZ;
    float* ws0 = (float*)d_ws;           // decomposed -> sem -> h
    float* ws1 = ws0 + N1;               // z1 / p1 -> fusion hidden
    float* ws2 = ws1 + N1;               // z / p
    float* partials = ws2 + N1;          // S_SZ * 256 floats

    const dim3 blk(256);
    const size_t HH = (size_t)H_SZ * H_SZ;

    // 1) decomposed = sr @ Wd : (4096x1024)@(1024x7168) -> ws0
    gemm_wmma_f32<false, false><<<dim3((S_SZ * H_SZ) / 128, B_SZ / 64, 1), blk, 0, stream>>>(
        sr, H_SZ, 0, Wd, 0, nullptr, 0, ws0, S_SZ * H_SZ, 0, S_SZ * H_SZ, H_SZ);

    // 2) sem = gelu(layer_norm(decomposed)) in place
    ln_gelu_kernel<<<B_SZ * S_SZ, blk, 0, stream>>>(ws0, gam, bet);

    // 3) h = gram_schmidt(sem) in place
    gram_schmidt_kernel<<<B_SZ, blk, 0, stream>>>(ws0);

    // 4) z1 = relu(h @ proj_w1[s] + proj_b1[s]) -> ws1
    gemm_wmma_f32<true, true><<<dim3(H_SZ / 128, B_SZ / 64, S_SZ), blk, 0, stream>>>(
        ws0, S_SZ * H_SZ, H_SZ, pw1, HH, pb1, H_SZ, ws1, S_SZ * H_SZ, H_SZ, H_SZ, H_SZ);
    // 5) z = z1 @ proj_w2[s] + proj_b2[s] -> ws2
    gemm_wmma_f32<true, false><<<dim3(H_SZ / 128, B_SZ / 64, S_SZ), blk, 0, stream>>>(
        ws1, S_SZ * H_SZ, H_SZ, pw2, HH, pb2, H_SZ, ws2, S_SZ * H_SZ, H_SZ, H_SZ, H_SZ);
    // 6) p1 = relu(z @ pred_w1[s] + pred_b1[s]) -> ws1
    gemm_wmma_f32<true, true><<<dim3(H_SZ / 128, B_SZ / 64, S_SZ), blk, 0, stream>>>(
        ws2, S_SZ * H_SZ, H_SZ, qw1, HH, qb1, H_SZ, ws1, S_SZ * H_SZ, H_SZ, H_SZ, H_SZ);
    // 7) p = p1 @ pred_w2[s] + pred_b2[s] -> ws2
    gemm_wmma_f32<true, false><<<dim3(H_SZ / 128, B_SZ / 64, S_SZ), blk, 0, stream>>>(
        ws1, S_SZ * H_SZ, H_SZ, qw2, HH, qb2, H_SZ, ws2, S_SZ * H_SZ, H_SZ, H_SZ, H_SZ);

    // 8-9) spr_losses / total -> out[B*H ..]
    loss_partial_kernel<<<dim3(256, S_SZ), blk, 0, stream>>>(ws2, ws0, partials);
    loss_final_kernel<<<1, 32, 0, stream>>>(partials, sw, out + (size_t)B_SZ * H_SZ);

    // 10) hidden = relu(p_flat @ fusion_w1 + fusion_b1) : (4096x7168)@(7168x2048) -> ws1
    gemm_wmma_f32<true, true><<<dim3(2048 / 128, B_SZ / 64, 1), blk, 0, stream>>>(
        ws2, S_SZ * H_SZ, 0, fw1, 0, fb1, 0, ws1, 2048, 0, 2048, S_SZ * H_SZ);
    // 11) final = hidden @ fusion_w2 + fusion_b2 : (4096x2048)@(2048x1024) -> out
    gemm_wmma_f32<true, false><<<dim3(H_SZ / 128, B_SZ / 64, 1), blk, 0, stream>>>(
        ws1, 2048, 0, fw2, 0, fb2, 0, out, H_SZ, 0, H_SZ, 2048);
}